// ImgtoClass_Metric_12927851560966
// MI455X (gfx1250) — compile-verified
//
#include <hip/hip_runtime.h>

typedef float v2f __attribute__((ext_vector_type(2)));
typedef float v8f __attribute__((ext_vector_type(8)));

// -----------------------------------------------------------------------------
// Kernel 1: normalized row-sum.  in: [n_rows, h_w, 64] f32 (row-major, C=64
// contiguous).  out: [grid, 64] f32, rows >= n_rows written as zeros (pad).
// One workgroup (256 threads = 8 wave32) per output row.  Each wave handles
// one descriptor j per iteration: lane holds C = {2*lane, 2*lane+1} as a
// float2 (coalesced 256B per wave), sum-of-squares via 5-step shuffle-xor
// wave reduction, scale by 1/sqrt, accumulate.  Cross-wave reduce in LDS.
// -----------------------------------------------------------------------------
__global__ void __launch_bounds__(256)
norm_sum_kernel(const float* __restrict__ in, float* __restrict__ out,
                int n_rows, int h_w) {
  const int i    = blockIdx.x;
  const int lane = threadIdx.x & 31;
  const int wave = threadIdx.x >> 5;

  __shared__ float red[8][64];

  float ax = 0.0f, ay = 0.0f;
  if (i < n_rows) {
    const float* base = in + (size_t)i * h_w * 64;
    for (int j = wave; j < h_w; j += 8) {
      const float2 v = reinterpret_cast<const float2*>(base + j * 64)[lane];
      float ss = v.x * v.x + v.y * v.y;
#pragma unroll
      for (int m = 16; m >= 1; m >>= 1) ss += __shfl_xor(ss, m, 32);
      const float r = 1.0f / sqrtf(ss);  // exact-norm division per reference
      ax = fmaf(v.x, r, ax);
      ay = fmaf(v.y, r, ay);
    }
  }
  red[wave][2 * lane]     = ax;
  red[wave][2 * lane + 1] = ay;
  __syncthreads();
  if (threadIdx.x < 64) {
    float s = 0.0f;
#pragma unroll
    for (int w = 0; w < 8; ++w) s += red[w][threadIdx.x];
    out[(size_t)i * 64 + threadIdx.x] = s;  // pad rows get 0 (ax/ay stayed 0)
  }
}

// -----------------------------------------------------------------------------
// Kernel 2: out[M,N] = Q[Mp,64] * S[Np,64]^T via V_WMMA_F32_16X16X4_F32.
// Q is zero-padded to Mp=80 rows, S to Np=32 rows, so every WMMA operand load
// is in-bounds and EXEC stays all-1s (WMMA requirement).  One wave per 16x16
// D tile; 5x2 = 10 tiles; K=64 -> 16 wmma steps, accumulated in the v8f C.
//
// A-frag (16x4 f32, ISA 7.12.2): lanes 0-15 -> M=lane, K={0,1}; lanes 16-31 ->
// M=lane-16, K={2,3}.  B-frag (4x16) mirrors with N across lanes.  D: VGPR v
// holds M=v (lanes 0-15) / M=v+8 (lanes 16-31), N=lane&15.
// -----------------------------------------------------------------------------
__global__ void __launch_bounds__(32)
wmma_gemm_f32_kernel(const float* __restrict__ Q, const float* __restrict__ S,
                     float* __restrict__ out, int M, int N) {
  const int lane  = threadIdx.x & 31;
  const int mt    = blockIdx.x % 5;       // row-tile  (80 / 16)
  const int nt    = blockIdx.x / 5;       // col-tile  (32 / 16)
  const int mn    = lane & 15;
  const int khalf = (lane >> 4) * 2;      // 0 or 2

  const float* qrow = Q + (size_t)(mt * 16 + mn) * 64;
  const float* srow = S + (size_t)(nt * 16 + mn) * 64;

  v8f c = {};
#pragma unroll
  for (int k0 = 0; k0 < 64; k0 += 4) {
    const float2 af = *reinterpret_cast<const float2*>(qrow + k0 + khalf);
    const float2 bf = *reinterpret_cast<const float2*>(srow + k0 + khalf);
    v2f a; a.x = af.x; a.y = af.y;
    v2f b; b.x = bf.x; b.y = bf.y;
    // 8 args: (neg_a, A, neg_b, B, c_mod, C, reuse_a, reuse_b)
    c = __builtin_amdgcn_wmma_f32_16x16x4_f32(false, a, false, b,
                                              (short)0, c, false, false);
  }

#pragma unroll
  for (int v = 0; v < 8; ++v) {
    const int row = mt * 16 + ((lane < 16) ? v : v + 8);
    const int col = nt * 16 + mn;
    if (row < M && col < N) out[(size_t)row * N + col] = c[v];
  }
}

// -----------------------------------------------------------------------------
// Launch: x1 [75,441,64] f32, x2 [1,5,5,441,64] f32 (domain 0 == whole buffer,
// reshaped to [25,441,64]).  d_ws layout: Qbar 80x64 f32 | Sbar 32x64 f32
// (28 KB, pads zeroed by kernel 1 since the harness poisons ws with 0xAA).
// d_out: 75*25 f32.
// -----------------------------------------------------------------------------
extern "C" void kernel_launch(void* const* d_in, const int* in_sizes, int n_in,
                              void* d_out, int out_size, void* d_ws, size_t ws_size,
                              hipStream_t stream) {
  const float* x1 = (const float*)d_in[0];
  const float* x2 = (const float*)d_in[1];
  float* out  = (float*)d_out;
  float* qbar = (float*)d_ws;        // [80][64]
  float* sbar = qbar + 80 * 64;      // [32][64]

  norm_sum_kernel<<<80, 256, 0, stream>>>(x1, qbar, 75, 441);
  norm_sum_kernel<<<32, 256, 0, stream>>>(x2, sbar, 25, 441);
  wmma_gemm_f32_kernel<<<10, 32, 0, stream>>>(qbar, sbar, out, 75, 25);
}